// TurboQuantValue_59639915872719
// MI455X (gfx1250) — compile-verified
//
#include <hip/hip_runtime.h>
#include <hip/hip_bf16.h>

// TurboQuantValue: per-128-element-row asymmetric Lloyd-Max quantize + dequant.
// 4 rows per wave tile, 8 lanes per row, 16 elems/lane. Async global->LDS
// double-buffered staging (padded LDS layout => conflict-free b128 reads),
// 3-step wave32 shuffle reductions, NT stores. HBM-bound target: ~22us.

#define TQ_DIM 128
#define TQ_TILE_ROWS 4
#define TQ_TILES_PER_WAVE 4      // 16 rows per wave
#define TQ_WAVES_PER_BLOCK 8
#define TQ_THREADS (TQ_WAVES_PER_BLOCK * 32)
#define TQ_ROW_F4 33             // 528B padded row stride in LDS (32 data + 1 pad)
#define TQ_TILE_F4 (TQ_TILE_ROWS * TQ_ROW_F4)  // 132 f4 = 2112B per tile buffer

typedef float f4 __attribute__((ext_vector_type(4)));
typedef int v4i __attribute__((ext_vector_type(4)));
typedef __attribute__((address_space(1))) v4i gv4i;  // global int4
typedef __attribute__((address_space(3))) v4i lv4i;  // LDS int4

#if defined(__has_builtin)
#if __has_builtin(__builtin_amdgcn_global_load_async_to_lds_b128) && \
    __has_builtin(__builtin_amdgcn_s_wait_asynccnt)
#define TQ_ASYNC_BUILTIN 1
#endif
#endif

// One 16B async copy global -> LDS (per lane).
__device__ __forceinline__ void tq_async_load16(const float* g, f4* l) {
#if defined(TQ_ASYNC_BUILTIN)
  __builtin_amdgcn_global_load_async_to_lds_b128((gv4i*)(void*)g,
                                                 (lv4i*)(void*)l, 0, 0);
#else
  unsigned lds_off = (unsigned)(unsigned long long)(
      (__attribute__((address_space(3))) void*)l);
  unsigned long long ga = (unsigned long long)g;
  asm volatile("global_load_async_to_lds_b128 %0, %1, off"
               :: "v"(lds_off), "v"(ga)
               : "memory");
#endif
}

__device__ __forceinline__ void tq_wait_async_le4() {
#if defined(TQ_ASYNC_BUILTIN)
  __builtin_amdgcn_s_wait_asynccnt(4);
#else
  asm volatile("s_wait_asynccnt 0x4" ::: "memory");
#endif
  asm volatile("" ::: "memory");
}

__device__ __forceinline__ void tq_wait_async_le0() {
#if defined(TQ_ASYNC_BUILTIN)
  __builtin_amdgcn_s_wait_asynccnt(0);
#else
  asm volatile("s_wait_asynccnt 0x0" ::: "memory");
#endif
  asm volatile("" ::: "memory");
}

// Stage one 4-row tile (2KB) into a padded LDS buffer: async instr i copies
// row (t*4+i): global bytes [16*lane,16*lane+16) -> LDS f4 slot 33*i + lane.
__device__ __forceinline__ void tq_stage_tile(const float* x, long long nrows,
                                              long long t, f4* buf, int lane) {
#pragma unroll
  for (int i = 0; i < TQ_TILE_ROWS; ++i) {
    long long r = t * TQ_TILE_ROWS + i;
    if (r >= nrows) r = nrows - 1;  // tail: duplicate last row (stores predicated)
    tq_async_load16(x + r * TQ_DIM + (long long)lane * 4,
                    buf + TQ_ROW_F4 * i + lane);
  }
}

// Reductions over the 8-lane row group (lanes with equal l>>3): masks 1,2,4.
__device__ __forceinline__ float tq_grp_min(float v) {
#pragma unroll
  for (int m = 1; m <= 4; m <<= 1) v = fminf(v, __shfl_xor(v, m, 32));
  return v;
}
__device__ __forceinline__ float tq_grp_sum(float v) {
#pragma unroll
  for (int m = 1; m <= 4; m <<= 1) v += __shfl_xor(v, m, 32);
  return v;
}

__global__ __launch_bounds__(TQ_THREADS) void TurboQuantValue_59639915872719_kernel(
    const float* __restrict__ x, const float* __restrict__ cbg,
    float* __restrict__ out, long long nrows) {
  __shared__ f4 smem[TQ_WAVES_PER_BLOCK * 2 * TQ_TILE_F4];

  const int lane = threadIdx.x & 31;
  const int wv   = threadIdx.x >> 5;
  const int r    = lane >> 3;   // row within tile handled by this lane
  const int k    = lane & 7;    // 64B segment within the row

  const long long ntiles = (nrows + TQ_TILE_ROWS - 1) / TQ_TILE_ROWS;
  const long long wid = (long long)blockIdx.x * TQ_WAVES_PER_BLOCK + wv;
  long long tile = wid * TQ_TILES_PER_WAVE;
  long long tend = tile + TQ_TILES_PER_WAVE;
  if (tend > ntiles) tend = ntiles;
  if (tile >= tend) return;

  // Codebook + midpoint boundaries: uniform -> SGPRs (scalar float ALU).
  float cb[16];
#pragma unroll
  for (int j = 0; j < 16; ++j) cb[j] = cbg[j];
  float bnd[15];
#pragma unroll
  for (int j = 0; j < 15; ++j) bnd[j] = 0.5f * (cb[j] + cb[j + 1]);
  // x_unit >= 0 always (xc = x - min >= 0). If boundaries 0..6 are negative
  // (sorted => checking bnd[6] suffices) those compares are constant-true and
  // the scan starts at cb[7]. Uniform branch, no divergence.
  const bool skip_neg = (bnd[6] < 0.0f);

  f4* buf0 = &smem[(wv * 2 + 0) * TQ_TILE_F4];
  f4* buf1 = &smem[(wv * 2 + 1) * TQ_TILE_F4];

  // Prologue: stage first tile (ASYNCcnt = 4).
  tq_stage_tile(x, nrows, tile, buf0, lane);
  int cur = 0;

  for (; tile < tend; ++tile) {
    const bool has_next = (tile + 1) < tend;
    f4* bcur = cur ? buf1 : buf0;
    f4* bnxt = cur ? buf0 : buf1;
    if (has_next) {
      tq_stage_tile(x, nrows, tile + 1, bnxt, lane);  // ASYNCcnt -> 8
      tq_wait_async_le4();  // current tile's 4 copies done (in-order retire)
    } else {
      tq_wait_async_le0();
    }

    // Conflict-free b128 reads: dword bank = (132r + 16k + 4i) % 64, every
    // bank hit exactly twice per instruction (the wave32 b128 minimum).
    float xv[16];
#pragma unroll
    for (int i = 0; i < 4; ++i) {
      f4 a = bcur[TQ_ROW_F4 * r + 4 * k + i];
      xv[4 * i + 0] = a.x; xv[4 * i + 1] = a.y;
      xv[4 * i + 2] = a.z; xv[4 * i + 3] = a.w;
    }

    // --- zero point: row min (pairwise tree + 3-step group reduce) ---
    float mm[8];
#pragma unroll
    for (int j = 0; j < 8; ++j) mm[j] = fminf(xv[j], xv[j + 8]);
#pragma unroll
    for (int s = 4; s >= 1; s >>= 1)
#pragma unroll
      for (int j = 0; j < 4; ++j)
        if (j < s) mm[j] = fminf(mm[j], mm[j + s]);
    const float vmin = tq_grp_min(mm[0]);

    // --- centered values + RMS scale ---
    float c[16];
    float ssa = 0.0f, ssb = 0.0f;
#pragma unroll
    for (int j = 0; j < 16; j += 2) {
      c[j]     = xv[j] - vmin;
      c[j + 1] = xv[j + 1] - vmin;
      ssa = fmaf(c[j], c[j], ssa);
      ssb = fmaf(c[j + 1], c[j + 1], ssb);
    }
    const float ss = tq_grp_sum(ssa + ssb);
    const float scale = __builtin_sqrtf(ss) / 11.313708498984760390f;  // sqrt(128)
    const float inv = 1.0f / (scale + 1e-10f);

    // --- nearest codeword (searchsorted side='left') + gamma sums ---
    float rr[16];
    float numa = 0.0f, numb = 0.0f, dena = 0.0f, denb = 0.0f;
    const int j0 = skip_neg ? 7 : 0;
    const float rinit = skip_neg ? cb[7] : cb[0];
#pragma unroll
    for (int j = 0; j < 16; ++j) {
      const float u = c[j] * inv;
      float rv = rinit;
      if (skip_neg) {
#pragma unroll
        for (int b = 7; b < 15; ++b) rv = (u > bnd[b]) ? cb[b + 1] : rv;
      } else {
#pragma unroll
        for (int b = 0; b < 15; ++b) rv = (u > bnd[b]) ? cb[b + 1] : rv;
      }
      rr[j] = rv;
      if (j & 1) { numb = fmaf(c[j], rv, numb); denb = fmaf(rv, rv, denb); }
      else       { numa = fmaf(c[j], rv, numa); dena = fmaf(rv, rv, dena); }
    }
    (void)j0;
    const float num = tq_grp_sum(numa + numb);
    const float den = tq_grp_sum(dena + denb);
    const float g = num / (den + 1e-10f);

    // --- dequantize + streaming stores (output never re-read) ---
    const long long orow = tile * TQ_TILE_ROWS + r;
    if (orow < nrows) {
#pragma unroll
      for (int i = 0; i < 4; ++i) {
        f4 o;
        o.x = fmaf(rr[4 * i + 0], g, vmin);
        o.y = fmaf(rr[4 * i + 1], g, vmin);
        o.z = fmaf(rr[4 * i + 2], g, vmin);
        o.w = fmaf(rr[4 * i + 3], g, vmin);
        __builtin_nontemporal_store(o, (f4*)out + orow * 32 + 4 * k + i);
      }
    }
    cur ^= 1;
  }
}

extern "C" void kernel_launch(void* const* d_in, const int* in_sizes, int n_in,
                              void* d_out, int out_size, void* d_ws, size_t ws_size,
                              hipStream_t stream) {
  const float* x   = (const float*)d_in[0];
  const float* cbk = (const float*)d_in[1];
  float* out = (float*)d_out;

  const long long n      = (long long)in_sizes[0];
  const long long nrows  = n / TQ_DIM;
  const long long ntiles = (nrows + TQ_TILE_ROWS - 1) / TQ_TILE_ROWS;
  const long long tiles_per_block =
      (long long)TQ_TILES_PER_WAVE * TQ_WAVES_PER_BLOCK;
  const int blocks = (int)((ntiles + tiles_per_block - 1) / tiles_per_block);

  TurboQuantValue_59639915872719_kernel<<<blocks, TQ_THREADS, 0, stream>>>(
      x, cbk, out, nrows);
}